// WeSpeR_MD_model_34497177321724
// MI455X (gfx1250) — compile-verified
//
#include <hip/hip_runtime.h>
#include <math.h>

#define NXP    8192
#define PK     128
#define CASP   0.5f
#define NITER  40
#define INV_PI 0.3183098861837907f

// Butterfly sum+broadcast within each 16-lane group of a wave32
// (xor masks < 16 never cross 16-aligned groups). Lowers to ds_swizzle + v_add.
__device__ __forceinline__ float group_sum16(float v) {
  v += __shfl_xor(v, 8, 32);
  v += __shfl_xor(v, 4, 32);
  v += __shfl_xor(v, 2, 32);
  v += __shfl_xor(v, 1, 32);
  return v;
}

__global__ __launch_bounds__(256)
void wesper_fixed_point_kernel(const float* __restrict__ x,
                               const float* __restrict__ d,
                               const float* __restrict__ wd,
                               const float* __restrict__ tau,
                               const float* __restrict__ wt,
                               float*  __restrict__ out_f,
                               float2* __restrict__ out_m,
                               float2* __restrict__ out_X) {
  __shared__ float s_d[PK], s_wd[PK], s_tau[PK], s_wt[PK];

  const int tid  = threadIdx.x;
  const int lane = tid & 31;
  const int wave = tid >> 5;
  const int sub  = lane >> 4;   // which of the 2 grid points this wave handles
  const int sl   = lane & 15;   // lane within the 16-lane point group

  // ---- Stage the four 128-float tables into LDS via the CDNA5 async path.
  // Waves 0..3 each DMA one table: 32 lanes x b128 = 512 B = one table.
  if (wave < 4) {
    const float* gsrc = (wave == 0) ? d : (wave == 1) ? wd : (wave == 2) ? tau : wt;
    float* ldst       = (wave == 0) ? s_d : (wave == 1) ? s_wd : (wave == 2) ? s_tau : s_wt;
    unsigned int       lds_off = (unsigned int)(unsigned long long)(&ldst[lane * 4]);
    unsigned long long gaddr   = (unsigned long long)(gsrc + lane * 4);
    asm volatile("global_load_async_to_lds_b128 %0, %1, off"
                 :: "v"(lds_off), "v"(gaddr) : "memory");
    asm volatile("s_wait_asynccnt 0" ::: "memory");
  }
  __syncthreads();

  // ---- Hoist per-lane constants: lane owns terms k = sl + 16u, u = 0..7.
  float tau_r[8], wtt_r[8], wt_r[8], cd_r[8], wdd_r[8];
  float wdd_part = 0.0f;
#pragma unroll
  for (int u = 0; u < 8; ++u) {
    const int k     = sl + 16 * u;
    const float tk  = s_tau[k];
    const float wk  = s_wt[k];
    const float dj  = s_d[k];
    const float wdj = s_wd[k];
    tau_r[u] = tk;
    wtt_r[u] = wk * tk;     // wt_k * tau_k
    wt_r[u]  = wk;
    cd_r[u]  = CASP * dj;   // c * d_j
    wdd_r[u] = wdj * dj;    // wd_j * d_j
    wdd_part += wdd_r[u];
  }
  const float m0r = group_sum16(wdd_part);   // sum(wd*d) = Re(m0)

  const int ix = (int)blockIdx.x * 16 + wave * 2 + sub;  // 16-lane group per point
  const float xv = x[ix];

  // ---- 40 damped fixed-point iterations on the Im<0 branch.
  float mr = m0r, mi = -1.0f;
#pragma unroll 1
  for (int it = 0; it < NITER; ++it) {
    // S(m) = sum_k wt_k*tau_k / (tau_k*m - x)
    float Sr_p = 0.0f, Si_p = 0.0f;
#pragma unroll
    for (int u = 0; u < 8; ++u) {
      const float dr  = __builtin_fmaf(tau_r[u], mr, -xv);
      const float di  = tau_r[u] * mi;
      const float inv = __builtin_amdgcn_rcpf(__builtin_fmaf(dr, dr, di * di));
      const float t   = wtt_r[u] * inv;
      Sr_p = __builtin_fmaf(t,  dr, Sr_p);
      Si_p = __builtin_fmaf(-t, di, Si_p);
    }
    const float Sr = group_sum16(Sr_p);
    const float Si = group_sum16(Si_p);

    // F = sum_j wd_j*d_j / (1 + c*d_j*S)
    float Fr_p = 0.0f, Fi_p = 0.0f;
#pragma unroll
    for (int u = 0; u < 8; ++u) {
      const float ar  = __builtin_fmaf(cd_r[u], Sr, 1.0f);
      const float ai  = cd_r[u] * Si;
      const float inv = __builtin_amdgcn_rcpf(__builtin_fmaf(ar, ar, ai * ai));
      const float t   = wdd_r[u] * inv;
      Fr_p = __builtin_fmaf(t,  ar, Fr_p);
      Fi_p = __builtin_fmaf(-t, ai, Fi_p);
    }
    const float Fr = group_sum16(Fr_p);
    const float Fi = fminf(group_sum16(Fi_p), -1e-10f);  // stay on Im<0 branch

    mr = 0.5f * (mr + Fr);   // damping
    mi = 0.5f * (mi + Fi);
  }

  // ---- Stieltjes transform m(x) = sum_k wt_k / (tau_k*m_tilde - x)
  float Mr_p = 0.0f, Mi_p = 0.0f;
#pragma unroll
  for (int u = 0; u < 8; ++u) {
    const float dr  = __builtin_fmaf(tau_r[u], mr, -xv);
    const float di  = tau_r[u] * mi;
    const float inv = __builtin_amdgcn_rcpf(__builtin_fmaf(dr, dr, di * di));
    const float t   = wt_r[u] * inv;
    Mr_p = __builtin_fmaf(t,  dr, Mr_p);
    Mi_p = __builtin_fmaf(-t, di, Mi_p);
  }
  const float Mr = group_sum16(Mr_p);
  const float Mi = group_sum16(Mi_p);

  if (sl == 0) {
    out_f[ix] = fabsf(Mi) * INV_PI;
    out_m[ix] = make_float2(Mr, Mi);
    const float invx = __builtin_amdgcn_rcpf(xv);
    out_X[ix] = make_float2(-mr * invx, -mi * invx);   // X = -m_tilde / x
  }
}

extern "C" void kernel_launch(void* const* d_in, const int* in_sizes, int n_in,
                              void* d_out, int out_size, void* d_ws, size_t ws_size,
                              hipStream_t stream) {
  (void)in_sizes; (void)n_in; (void)out_size; (void)d_ws; (void)ws_size;
  // setup_inputs order: x[8192], d[128], wd[128], tau[128], wt[128]
  const float* x   = (const float*)d_in[0];
  const float* d   = (const float*)d_in[1];
  const float* wd  = (const float*)d_in[2];
  const float* tau = (const float*)d_in[3];
  const float* wt  = (const float*)d_in[4];

  float*  out   = (float*)d_out;
  float*  out_f = out;                               // f: 8192 f32
  float2* out_m = (float2*)(out + NXP);              // m: 8192 complex64
  float2* out_X = (float2*)(out + NXP + 2 * NXP);    // X: 8192 complex64

  dim3 grid(NXP / 16), block(256);   // 8 waves/block, 2 points/wave (16 lanes each)
  hipLaunchKernelGGL(wesper_fixed_point_kernel, grid, block, 0, stream,
                     x, d, wd, tau, wt, out_f, out_m, out_X);
}